// TSpatialConvolution_12128987644193
// MI455X (gfx1250) — compile-verified
//
#include <hip/hip_runtime.h>

typedef float v2f __attribute__((ext_vector_type(2)));
typedef float v8f __attribute__((ext_vector_type(8)));

#define N_NODES 8192
#define KSL 8                      // trailing slice dim of S/X
#define F_INF 32
#define F_OUTF 32
#define KTOT (N_NODES * KSL)       // 65536 contraction length
#define WAVES_PER_BLOCK 8
#define K_PER_WAVE (KTOT / WAVES_PER_BLOCK)   // 8192
#define STEPS_PER_WAVE (K_PER_WAVE / 4)       // 2048 wmma k-steps per wave

// ---------------------------------------------------------------------------
// Pre-pass: pack X (m, f, k) into the exact per-lane WMMA B-fragment stream.
// Layout: PB[((t*2 + ftile)*32 + lane)*2 + j], where t = k-step (K=4 chunk),
// lane 0-15 hold B[K=0/1, f] in j=0/1, lanes 16-31 hold B[K=2/3, f].
// B[(m*KSL+kk), f] = X[m, f, kk].
// ---------------------------------------------------------------------------
__global__ void pack_b_kernel(const float* __restrict__ X, float* __restrict__ PB) {
    int idx = blockIdx.x * blockDim.x + threadIdx.x;     // 0 .. 2,097,151
    int j     = idx & 1;
    int lane  = (idx >> 1) & 31;
    int ftile = (idx >> 6) & 1;
    int t     = idx >> 7;
    int k = t * 4 + ((lane >> 4) << 1) + j;              // contraction index
    int f = (lane & 15) + (ftile << 4);
    int m  = k >> 3;
    int kk = k & 7;
    PB[idx] = X[((size_t)m * F_INF + f) * KSL + kk];
}

// ---------------------------------------------------------------------------
// Main: one block per 16-row n-tile. 8 waves split the K=65536 contraction.
// Each wave: 2048 steps of 2x v_wmma_f32_16x16x4_f32 (A reused for both
// f-tiles). Partials reduced via LDS, then 32x32 W epilogue.
// ---------------------------------------------------------------------------
__global__ __launch_bounds__(256) void tconv_wmma_kernel(
    const float* __restrict__ S,   // viewed as A[8192][65536] row-major
    const float* __restrict__ PB,  // packed B fragment stream
    const float* __restrict__ W,   // [32][32]
    float* __restrict__ out) {     // [8192][32]
    __shared__ float lds[WAVES_PER_BLOCK * 512 + 512];

    const int tid  = threadIdx.x;
    const int lane = tid & 31;
    const int wave = tid >> 5;
    const int n0   = blockIdx.x * 16;

    // A fragment addressing: lane 0-15 -> rows 0-15, K=0/1; lanes 16-31 -> K=2/3
    const int row  = lane & 15;
    const int koff = (lane >> 4) << 1;

    const float* aptr = S + (size_t)(n0 + row) * KTOT + (size_t)wave * K_PER_WAVE + koff;
    const size_t t0 = (size_t)wave * STEPS_PER_WAVE;
    const float* bptr0 = PB + t0 * 128 + lane * 2;       // f-tile 0
    const float* bptr1 = bptr0 + 64;                     // f-tile 1

    v8f acc0 = {};
    v8f acc1 = {};

    #pragma unroll 2
    for (int t = 0; t < STEPS_PER_WAVE; ++t) {
        v2f a  = *(const v2f*)aptr;     // 8B/lane, 16B stride -> cacheline reuse
        v2f b0 = *(const v2f*)bptr0;    // contiguous packed fragment stream
        v2f b1 = *(const v2f*)bptr1;
        acc0 = __builtin_amdgcn_wmma_f32_16x16x4_f32(
            /*neg_a=*/false, a, /*neg_b=*/false, b0,
            /*c_mod=*/(short)0, acc0, /*reuse_a=*/false, /*reuse_b=*/false);
        acc1 = __builtin_amdgcn_wmma_f32_16x16x4_f32(
            false, a, false, b1, (short)0, acc1, false, false);
        aptr  += 4;     // K advances by 4 floats per step
        bptr0 += 128;   // 2 tiles * 32 lanes * 2 floats
        bptr1 += 128;
    }

    // Spill per-wave 16x32 partial tile to LDS using the C/D VGPR layout:
    // VGPR i holds row (i + 8*(lane>=16)), col (lane&15).
    {
        float* p = lds + wave * 512;
        const int rbase = (lane >> 4) << 3;
        const int col   = lane & 15;
        #pragma unroll
        for (int i = 0; i < 8; ++i) {
            p[(rbase + i) * 32 + col]      = acc0[i];
            p[(rbase + i) * 32 + 16 + col] = acc1[i];
        }
    }
    __syncthreads();

    // Cross-wave reduction: 512 tile elements, 2 per thread.
    for (int e = tid; e < 512; e += 256) {
        float s = 0.f;
        #pragma unroll
        for (int w = 0; w < WAVES_PER_BLOCK; ++w) s += lds[w * 512 + e];
        lds[WAVES_PER_BLOCK * 512 + e] = s;
    }
    __syncthreads();

    // Epilogue: out[n0+r, fo] = sum_f C[r,f] * W[f,fo]
    const float* Csum = lds + WAVES_PER_BLOCK * 512;
    for (int e = tid; e < 512; e += 256) {
        const int r  = e >> 5;
        const int fo = e & 31;
        float s = 0.f;
        #pragma unroll
        for (int f = 0; f < F_INF; ++f)
            s += Csum[r * 32 + f] * W[f * F_OUTF + fo];
        out[(size_t)(n0 + r) * F_OUTF + fo] = s;
    }
}

extern "C" void kernel_launch(void* const* d_in, const int* in_sizes, int n_in,
                              void* d_out, int out_size, void* d_ws, size_t ws_size,
                              hipStream_t stream) {
    (void)in_sizes; (void)n_in; (void)out_size; (void)ws_size;
    const float* S = (const float*)d_in[0];   // (8192, 8192, 8) fp32
    const float* X = (const float*)d_in[1];   // (8192, 32, 8)  fp32
    const float* W = (const float*)d_in[2];   // (32, 32)       fp32
    float* out = (float*)d_out;               // (8192, 32)     fp32
    float* PB  = (float*)d_ws;                // 8 MB packed B fragments

    const int pack_elems = (KTOT / 4) * 2 * 32 * 2;   // 2,097,152
    pack_b_kernel<<<pack_elems / 256, 256, 0, stream>>>(X, PB);
    tconv_wmma_kernel<<<N_NODES / 16, 256, 0, stream>>>(S, PB, W, out);
}